// PhysicsInformedLoss_979252543646
// MI455X (gfx1250) — compile-verified
//
#include <hip/hip_runtime.h>

// MI455X / gfx1250 physics-informed loss.
// Bandwidth-bound stencil (134 MB -> ~5.8us floor @ 23.3 TB/s).
// Main pass: coalesced b128 streaming loads, f32 VALU stencil math.
// Wave reduction: V_WMMA_F32_16X16X4_F32 ones-matrix trick (wave32).

typedef float v2f __attribute__((ext_vector_type(2)));
typedef float v8f __attribute__((ext_vector_type(8)));

namespace {
constexpr int kNX = 1024;
constexpr int kNY = 512;
constexpr double kDX = 0.26 / (kNX - 1);
constexpr double kDY = 0.12 / (kNY - 1);
constexpr float INV2DX = (float)(1.0 / (2.0 * kDX));
constexpr float INV2DY = (float)(1.0 / (2.0 * kDY));
constexpr float INVDX2 = (float)(1.0 / (kDX * kDX));
constexpr float INVDY2 = (float)(1.0 / (kDY * kDY));
constexpr float kNU   = 1.0e-4f;
constexpr float kUIN  = 0.1f;
constexpr int kAcc = 10;  // 10 global accumulators
}

// Sum a value across all 32 lanes of a wave (result broadcast to all lanes).
// Uses V_WMMA_F32_16X16X4_F32 with B == ones:
//   A[m][0] = lane m, A[m][2] = lane m+16, A[m][1] = A[m][3] = 0
//   => D[m][n] = acc[m] + acc[m+16]  (independent of n)
// Per-lane sum of the 8 D VGPRs gives half-sums; one shfl_xor(16) finishes.
__device__ __forceinline__ float wave_sum32(float v) {
#if __has_builtin(__builtin_amdgcn_wmma_f32_16x16x4_f32)
  v2f a; a[0] = v;    a[1] = 0.0f;
  v2f b; b[0] = 1.0f; b[1] = 1.0f;
  v8f c = {};
  v8f d = __builtin_amdgcn_wmma_f32_16x16x4_f32(
      /*neg_a=*/false, a, /*neg_b=*/false, b,
      /*c_mod=*/(short)0, c, /*reuse_a=*/false, /*reuse_b=*/false);
  float s = ((d[0] + d[1]) + (d[2] + d[3])) + ((d[4] + d[5]) + (d[6] + d[7]));
  s += __shfl_xor(s, 16, 32);
  return s;
#else
  #pragma unroll
  for (int o = 16; o > 0; o >>= 1) v += __shfl_xor(v, o, 32);
  return v;
#endif
}

__global__ void pinn_zero_ws(float* __restrict__ ws) {
  if (threadIdx.x < 16) ws[threadIdx.x] = 0.0f;
}

// Block: 32 x 8 (8 wave32s). Lane tx owns j = 4*(32*bx+tx) .. +3 (float4),
// ty/by select row i, bz selects batch image.
__global__ __launch_bounds__(256) void pinn_main(const float* __restrict__ yh,
                                                 const int* __restrict__ xin,
                                                 float* __restrict__ ws) {
  const int tx = threadIdx.x;
  const int ty = threadIdx.y;
  const int j0 = (blockIdx.x * 32 + tx) * 4;
  const int i  = blockIdx.y * 8 + ty;
  const int bb = blockIdx.z;

  const size_t plane = (size_t)kNX * kNY;
  const float* ux = yh + (size_t)bb * 3 * plane;
  const float* uy = ux + plane;
  const float* pp = uy + plane;
  const int*   rg = xin + (size_t)bb * 2 * plane + plane;  // x_in channel 1 only

  const int im = (i > 0) ? i - 1 : 0;
  const int ip = (i < kNX - 1) ? i + 1 : kNX - 1;
  const int rc = i  * kNY + j0;
  const int rm = im * kNY + j0;
  const int rp = ip * kNY + j0;
  const int jml = i * kNY + ((j0 > 0) ? j0 - 1 : 0);
  const int jpl = i * kNY + ((j0 + 4 < kNY) ? j0 + 4 : kNY - 1);

  // ---- streaming loads (coalesced b128) ----
  const float4 uxc4 = *(const float4*)(ux + rc);
  const float4 uxm4 = *(const float4*)(ux + rm);
  const float4 uxp4 = *(const float4*)(ux + rp);
  const float  uxL  = ux[jml], uxR = ux[jpl];

  const float4 uyc4 = *(const float4*)(uy + rc);
  const float4 uym4 = *(const float4*)(uy + rm);
  const float4 uyp4 = *(const float4*)(uy + rp);
  const float  uyL  = uy[jml], uyR = uy[jpl];

  const float4 pc4 = *(const float4*)(pp + rc);
  const float4 pm4 = *(const float4*)(pp + rm);
  const float4 pp4 = *(const float4*)(pp + rp);
  const float  pL  = pp[jml], pR = pp[jpl];

  const int4 r4 = *(const int4*)(rg + rc);

  const float uxc[4]  = {uxc4.x, uxc4.y, uxc4.z, uxc4.w};
  const float uxxm[4] = {uxm4.x, uxm4.y, uxm4.z, uxm4.w};
  const float uxxp[4] = {uxp4.x, uxp4.y, uxp4.z, uxp4.w};
  const float uxym[4] = {uxL, uxc4.x, uxc4.y, uxc4.z};
  const float uxyp[4] = {uxc4.y, uxc4.z, uxc4.w, uxR};

  const float uyc[4]  = {uyc4.x, uyc4.y, uyc4.z, uyc4.w};
  const float uyxm[4] = {uym4.x, uym4.y, uym4.z, uym4.w};
  const float uyxp[4] = {uyp4.x, uyp4.y, uyp4.z, uyp4.w};
  const float uyym[4] = {uyL, uyc4.x, uyc4.y, uyc4.z};
  const float uyyp[4] = {uyc4.y, uyc4.z, uyc4.w, uyR};

  const float pxm[4] = {pm4.x, pm4.y, pm4.z, pm4.w};
  const float pxp[4] = {pp4.x, pp4.y, pp4.z, pp4.w};
  const float pym[4] = {pL, pc4.x, pc4.y, pc4.z};
  const float pyp[4] = {pc4.y, pc4.z, pc4.w, pR};

  const int rv[4] = {r4.x, r4.y, r4.z, r4.w};

  float acc[kAcc];
  #pragma unroll
  for (int c = 0; c < kAcc; ++c) acc[c] = 0.0f;

  const bool i_in = (i >= 1) && (i <= kNX - 2);

  #pragma unroll
  for (int k = 0; k < 4; ++k) {
    const float dudx = (uxxp[k] - uxxm[k]) * INV2DX;
    const float dudy = (uxyp[k] - uxym[k]) * INV2DY;
    const float dvdx = (uyxp[k] - uyxm[k]) * INV2DX;
    const float dvdy = (uyyp[k] - uyym[k]) * INV2DY;
    const float dpdx = (pxp[k] - pxm[k]) * INV2DX;
    const float dpdy = (pyp[k] - pym[k]) * INV2DY;

    const float lap_u = (uxxp[k] - 2.0f * uxc[k] + uxxm[k]) * INVDX2 +
                        (uxyp[k] - 2.0f * uxc[k] + uxym[k]) * INVDY2;
    const float lap_v = (uyxp[k] - 2.0f * uyc[k] + uyxm[k]) * INVDX2 +
                        (uyyp[k] - 2.0f * uyc[k] + uyym[k]) * INVDY2;

    const float divv = dudx + dvdy;
    const float rx = uxc[k] * dudx + uyc[k] * dudy + dpdx - kNU * lap_u;
    const float ry = uxc[k] * dvdx + uyc[k] * dvdy + dpdy - kNU * lap_v;

    const int r = rv[k];
    const int j = j0 + k;
    const float fint = (r >= 1 && i_in && j >= 1 && j <= kNY - 2) ? 1.0f : 0.0f;
    const float fns  = (r == 0 || r == 2) ? 1.0f : 0.0f;
    const float fin  = (r == 3) ? 1.0f : 0.0f;
    const float fou  = (r == 4) ? 1.0f : 0.0f;

    acc[0] += fint;
    acc[1] += fint * divv * divv;
    acc[2] += fint * rx * rx;
    acc[3] += fint * ry * ry;
    acc[4] += fns;
    acc[5] += fns * (uxc[k] * uxc[k] + uyc[k] * uyc[k]);
    acc[6] += fin;
    const float du = uxc[k] - kUIN;
    acc[7] += fin * (du * du + uyc[k] * uyc[k]);
    acc[8] += fou;
    acc[9] += fou * dudx * dudx;
  }

  // ---- wave32 reduction via WMMA (EXEC all-ones here: no divergence above) ----
  float wsum[kAcc];
  #pragma unroll
  for (int c = 0; c < kAcc; ++c) wsum[c] = wave_sum32(acc[c]);

  __shared__ float red[8][kAcc];
  if (tx == 0) {
    #pragma unroll
    for (int c = 0; c < kAcc; ++c) red[ty][c] = wsum[c];
  }
  __syncthreads();

  const int tid = ty * 32 + tx;
  if (tid < kAcc) {
    float s = 0.0f;
    #pragma unroll
    for (int w = 0; w < 8; ++w) s += red[w][tid];
    atomicAdd(&ws[tid], s);
  }
}

__global__ void pinn_finalize(const float* __restrict__ ws, float* __restrict__ out) {
  if (threadIdx.x == 0) {
    const float n_int = fmaxf(ws[0], 1.0f);
    const float l_cont = ws[1] / n_int;
    const float l_mom  = (ws[2] + ws[3]) / n_int;
    const float l_bc   = ws[5] / fmaxf(ws[4], 1.0f)
                       + ws[7] / fmaxf(ws[6], 1.0f)
                       + ws[9] / fmaxf(ws[8], 1.0f);
    out[0] = l_cont;
    out[1] = l_mom;
    out[2] = l_bc;
    out[3] = 1.0f * l_cont + 0.1f * l_mom + 1.0f * l_bc;
  }
}

extern "C" void kernel_launch(void* const* d_in, const int* in_sizes, int n_in,
                              void* d_out, int out_size, void* d_ws, size_t ws_size,
                              hipStream_t stream) {
  const float* yh  = (const float*)d_in[0];
  const int*   xin = (const int*)d_in[1];
  float* ws  = (float*)d_ws;
  float* out = (float*)d_out;

  const int batches = in_sizes[0] / (3 * kNX * kNY);  // 16

  pinn_zero_ws<<<1, 32, 0, stream>>>(ws);

  dim3 blk(32, 8, 1);
  dim3 grd(kNY / 128, kNX / 8, batches);
  pinn_main<<<grd, blk, 0, stream>>>(yh, xin, ws);

  pinn_finalize<<<1, 32, 0, stream>>>(ws, out);
}